// RandomAugList_60086592471445
// MI455X (gfx1250) — compile-verified
//
#include <hip/hip_runtime.h>

// MoE per-sample 3x3 SAME conv, fp32, implicit GEMM on V_WMMA_F32_16X16X4_F32.
//
// Roofline: 1.36 GFLOP vs 201 MB HBM traffic -> memory bound (~8.6 us at
// 23.3 TB/s). Strategy: keep fp32 end-to-end (bit-matches reference math),
// stream the image with coalesced constant-offset loads (tap re-reads hit
// WGP$/L2; 192 MB L2 nearly holds the whole problem), and put the MACs on the
// WMMA pipe so the VALU stays idle.
//
// GEMM per wave (wave32), one wave per image row:
//   D[16 px, 16 ch-slots] = sum_s A[16 px, 4] x B[4, 16] + C(bias),  s=0..6
//   K = 27 taps (ic*9+ky*3+kx) padded to 28; pad column of B is zero, so the
//   pad slot of A may read any *valid* address (0 * x == 0) -> no mask needed.
// Zero-padding cases:
//   - y edge (rows 0 and 511): wave-uniform -> masked slow path (2/512 waves)
//   - x edge (tiles 0 and 31): peeled, masked with v_cndmask (no branches)
//   - interior 30 tiles: fully unrolled, loads are base-VGPR + immediate
//     offset, zero address math, zero masks; first WMMA takes C = bias.

typedef float v2f __attribute__((ext_vector_type(2)));
typedef float v8f __attribute__((ext_vector_type(8)));

#define HDIM 512
#define WDIM 512
#define KSTEPS 7  // ceil(27/4)

__device__ __forceinline__ v8f wmma_step(const v2f& a, const v2f& b, const v8f& c) {
  return __builtin_amdgcn_wmma_f32_16x16x4_f32(false, a, false, b, (short)0, c,
                                               false, false);
}

// Masked tile: full bounds checks via selects (used on x-edge tiles and on
// the two y-edge rows only).
__device__ __forceinline__ v8f masked_tile(int x0, int y, int m, int hi,
                                           const float* __restrict__ imgb,
                                           const v2f* bfrag, const v8f& cinit) {
  v8f acc = cinit;
#pragma unroll
  for (int s = 0; s < KSTEPS; ++s) {
    v2f a;
#pragma unroll
    for (int v = 0; v < 2; ++v) {
      const int k  = 4 * s + 2 * hi + v;
      const int kc = (k < 27) ? k : 0;  // pad slot: B column is zero anyway
      const int ic = kc / 9;
      const int r  = kc - ic * 9;
      const int ky = r / 3;
      const int kx = r - ky * 3;
      const int yy  = y + ky - 1;
      const int yyc = min(max(yy, 0), HDIM - 1);
      const int xx  = x0 + m + kx - 1;
      const int xxc = min(max(xx, 0), WDIM - 1);
      const float val = imgb[((size_t)ic * HDIM + yyc) * WDIM + xxc];
      const bool ok = (yy >= 0) & (yy < HDIM) & (xx >= 0) & (xx < WDIM);
      a[v] = ok ? val : 0.0f;
    }
    acc = wmma_step(a, bfrag[s], acc);
  }
  return acc;
}

__global__ __launch_bounds__(256) void moe_conv3x3_wmma(
    const float* __restrict__ img,    // [B,3,H,W]
    const float* __restrict__ Wt,     // [E,3,3,3,3] (e,oc,ic,ky,kx)
    const float* __restrict__ bias,   // [E,3]
    const int*   __restrict__ alloc,  // [B]
    float* __restrict__ out)          // [B,3,H,W]
{
  const int lane = threadIdx.x & 31;
  const int wave = blockIdx.x * (blockDim.x >> 5) + (threadIdx.x >> 5);
  const int b  = wave >> 9;        // one wave per image row
  const int y  = wave & (HDIM - 1);
  const int m  = lane & 15;        // M: pixel within 16-wide tile
  const int hi = lane >> 4;        // half-wave selector
  const int oc = m;                // N: output-channel slot (0..2 useful)

  const int e = alloc[b];

  // B-matrix fragments: lane holds B[K = 4s + 2*hi + v, N = oc].
  // W flat index: e*81 + oc*27 + k  (k = ic*9 + ky*3 + kx).
  v2f bfrag[KSTEPS];
  {
    const float* we = Wt + e * 81;
#pragma unroll
    for (int s = 0; s < KSTEPS; ++s) {
#pragma unroll
      for (int v = 0; v < 2; ++v) {
        const int k = 4 * s + 2 * hi + v;
        float wv = 0.0f;
        if (oc < 3 && k < 27) wv = we[oc * 27 + k];
        bfrag[s][v] = wv;
      }
    }
  }

  // Bias folded into the accumulator init (C operand of the first WMMA):
  // D[M,N] needs +bias[N] for every M -> all 8 C VGPRs = bias[oc].
  const float bv = (oc < 3) ? bias[e * 3 + oc] : 0.0f;
  v8f cinit;
#pragma unroll
  for (int v = 0; v < 8; ++v) cinit[v] = bv;

  const float* imgb = img + (size_t)b * (3u * HDIM * WDIM);
  float* outb       = out + (size_t)b * (3u * HDIM * WDIM);

  // Store base: lane (oc<3) owns D[Mrow = v + 8*hi, N=oc] -> 8 consecutive x.
  const bool st_ok = (oc < 3);
  float* outrow = outb + ((size_t)(st_ok ? oc : 0) * HDIM + y) * WDIM + 8 * hi;

  if (y == 0 || y == HDIM - 1) {
    // Rare wave-uniform slow path: both image y-neighbors need masking.
#pragma unroll 1
    for (int xt = 0; xt < WDIM / 16; ++xt) {
      const int x0 = xt * 16;
      const v8f acc = masked_tile(x0, y, m, hi, imgb, bfrag, cinit);
      if (st_ok) {
        float* p = outrow + x0;
        *(float4*)(p)     = make_float4(acc[0], acc[1], acc[2], acc[3]);
        *(float4*)(p + 4) = make_float4(acc[4], acc[5], acc[6], acc[7]);
      }
    }
    return;
  }

  // ---- steady state: interior row ----
  // Per-tap base pointers (pad slot aliases tap 0: its B column is zero).
  const float* fb[KSTEPS][2];
#pragma unroll
  for (int s = 0; s < KSTEPS; ++s) {
#pragma unroll
    for (int v = 0; v < 2; ++v) {
      const int k  = 4 * s + 2 * hi + v;
      const int kc = (k < 27) ? k : 0;
      const int ic = kc / 9;
      const int r  = kc - ic * 9;
      const int ky = r / 3;
      const int kx = r - ky * 3;
      const int yy = y + ky - 1;  // always in [0, H-1] here
      fb[s][v] = imgb + ((size_t)ic * HDIM + yy) * WDIM + (m + kx - 1);
    }
  }

  // Peeled left edge tile (x = -1 possible).
  {
    const v8f acc = masked_tile(0, y, m, hi, imgb, bfrag, cinit);
    if (st_ok) {
      *(float4*)(outrow)     = make_float4(acc[0], acc[1], acc[2], acc[3]);
      *(float4*)(outrow + 4) = make_float4(acc[4], acc[5], acc[6], acc[7]);
    }
  }

  // Interior tiles: fully unrolled -> every load is base + immediate offset,
  // no masks, no address math; WMMA chain consumes loads as they land.
#pragma unroll
  for (int xt = 1; xt < WDIM / 16 - 1; ++xt) {
    const int x0 = xt * 16;
    v8f acc = cinit;
#pragma unroll
    for (int s = 0; s < KSTEPS; ++s) {
      v2f a;
      a[0] = fb[s][0][x0];
      a[1] = fb[s][1][x0];
      acc = wmma_step(a, bfrag[s], acc);
    }
    if (st_ok) {
      float* p = outrow + x0;
      *(float4*)(p)     = make_float4(acc[0], acc[1], acc[2], acc[3]);
      *(float4*)(p + 4) = make_float4(acc[4], acc[5], acc[6], acc[7]);
    }
  }

  // Peeled right edge tile (x = 512 possible).
  {
    const int x0 = WDIM - 16;
    const v8f acc = masked_tile(x0, y, m, hi, imgb, bfrag, cinit);
    if (st_ok) {
      float* p = outrow + x0;
      *(float4*)(p)     = make_float4(acc[0], acc[1], acc[2], acc[3]);
      *(float4*)(p + 4) = make_float4(acc[4], acc[5], acc[6], acc[7]);
    }
  }
}

extern "C" void kernel_launch(void* const* d_in, const int* in_sizes, int n_in,
                              void* d_out, int out_size, void* d_ws, size_t ws_size,
                              hipStream_t stream) {
  const float* img   = (const float*)d_in[0];
  const float* Wt    = (const float*)d_in[1];
  const float* bias  = (const float*)d_in[2];
  const int*   alloc = (const int*)d_in[3];
  float* out = (float*)d_out;

  const int B = in_sizes[3];      // number of samples (32)
  const int waves = B * HDIM;     // one wave per image row
  const int blocks = waves / 8;   // 8 wave32s per 256-thread block
  hipLaunchKernelGGL(moe_conv3x3_wmma, dim3(blocks), dim3(256), 0, stream,
                     img, Wt, bias, alloc, out);
}